// DynamicPositionalEmbedding_19859928776845
// MI455X (gfx1250) — compile-verified
//
#include <hip/hip_runtime.h>
#include <hip/hip_bf16.h>
#include <math.h>

// Problem constants (from reference setup_inputs)
#define BATCH 32
#define I_DIM 1024
#define J_DIM 2048
#define C_DIM 1024
#define K_DIM 2048   // 2*I_DIM, GEMM reduction dim
#define EPSN 1e-12f

typedef __attribute__((ext_vector_type(2))) float v2f;
typedef __attribute__((ext_vector_type(8))) float v8f;

// ---------------------------------------------------------------------------
// Kernel 1: per-batch statistics.
//   stats[b]      = mean(top16(h_normalized[b,:]))   (T)
//   stats[32+b]   = mean(bottom16(h_normalized[b,:])) (Btm)
//   stats[64+b]   = 1 / max(||f_rad[b,:]||, eps)      (inv_nr)
// One block (256 threads) per batch. h sorted with a bitonic sort in LDS;
// sorting raw values is valid since division by the (positive) norm is
// order-preserving — normalize the means afterward.
// ---------------------------------------------------------------------------
__global__ void dpe_stats_kernel(const float* __restrict__ f_rad,
                                 const float* __restrict__ f_histo,
                                 float* __restrict__ stats) {
  const int b   = blockIdx.x;
  const int tid = threadIdx.x;
  __shared__ float s[J_DIM];    // 8 KB of the 320 KB WGP LDS
  __shared__ float red[256];

  // Load histo row + sum of squares
  float ss = 0.f;
  for (int j = tid; j < J_DIM; j += 256) {
    float v = f_histo[b * J_DIM + j];
    s[j] = v;
    ss += v * v;
  }
  red[tid] = ss;
  __syncthreads();
  for (int off = 128; off > 0; off >>= 1) {
    if (tid < off) red[tid] += red[tid + off];
    __syncthreads();
  }
  const float nh = fmaxf(sqrtf(red[0]), EPSN);
  __syncthreads();

  // Bitonic sort ascending (2048 elements, 256 threads, 8 elems/thread/pass)
  for (int k = 2; k <= J_DIM; k <<= 1) {
    for (int j = k >> 1; j > 0; j >>= 1) {
      for (int i = tid; i < J_DIM; i += 256) {
        int ixj = i ^ j;
        if (ixj > i) {
          bool up = ((i & k) == 0);
          float a = s[i], c = s[ixj];
          if ((a > c) == up) { s[i] = c; s[ixj] = a; }
        }
      }
      __syncthreads();
    }
  }

  // Rad row norm
  float ssr = 0.f;
  for (int i = tid; i < I_DIM; i += 256) {
    float v = f_rad[b * I_DIM + i];
    ssr += v * v;
  }
  red[tid] = ssr;
  __syncthreads();
  for (int off = 128; off > 0; off >>= 1) {
    if (tid < off) red[tid] += red[tid + off];
    __syncthreads();
  }

  if (tid == 0) {
    float top = 0.f, bot = 0.f;
    for (int t = 0; t < 16; ++t) {
      bot += s[t];
      top += s[J_DIM - 16 + t];
    }
    const float nr = fmaxf(sqrtf(red[0]), EPSN);
    stats[b]      = (top * (1.0f / 16.0f)) / nh;  // T
    stats[32 + b] = (bot * (1.0f / 16.0f)) / nh;  // Btm
    stats[64 + b] = 1.0f / nr;                    // inv ||f_rad||
  }
}

// ---------------------------------------------------------------------------
// Kernel 2: build feat[32, 2048]:
//   feat[b, i]      = r>0 ?  r*T   : r*Btm   (pos_map)
//   feat[b, I+i]    = r>0 ? -r*Btm : -r*T    (neg_map)
// where r = f_rad[b,i] * inv_nr[b].
// ---------------------------------------------------------------------------
__global__ void dpe_feat_kernel(const float* __restrict__ f_rad,
                                const float* __restrict__ stats,
                                float* __restrict__ feat) {
  const int idx = blockIdx.x * 256 + threadIdx.x;   // BATCH * I_DIM threads
  const int b = idx >> 10;         // / I_DIM
  const int i = idx & (I_DIM - 1);
  const float T   = stats[b];
  const float Bt  = stats[32 + b];
  const float inr = stats[64 + b];
  const float r = f_rad[b * I_DIM + i] * inr;
  const float pos = (r > 0.f) ? r * T  : r * Bt;
  const float neg = (r > 0.f) ? -r * Bt : -r * T;
  feat[b * K_DIM + i]         = pos;
  feat[b * K_DIM + I_DIM + i] = neg;
}

// ---------------------------------------------------------------------------
// Kernel 3: emb[32,1024] = feat[32,2048] @ W[1024,2048]^T + bias + token*flag
// One wave (32 lanes) per 16-column tile of C (64 blocks). Each wave holds
// BOTH 16-row M-tiles (B=32) as two f32 16x16 accumulators, so the dominant
// stream (W, 8 MB) is read exactly once from HBM.
//
// WMMA f32 16x16x4 operand layout (ISA 7.12.2, 32-bit A 16x4):
//   lanes 0-15 : VGPR0=K0, VGPR1=K1 (M=lane)
//   lanes 16-31: VGPR0=K2, VGPR1=K3 (M=lane-16)
// -> each lane loads a float2 at (row, k + 2*(lane>=16)). B (=W^T tile,
// B[k][n] = W[n][k]) uses the mirrored layout, same float2 addressing on W.
// C/D: VGPR i -> row i (lanes 0-15) / row 8+i (lanes 16-31), col = lane&15.
// ---------------------------------------------------------------------------
__global__ void dpe_gemm_kernel(const float* __restrict__ feat,
                                const float* __restrict__ W,
                                const float* __restrict__ bias,
                                const float* __restrict__ token,
                                const unsigned char* __restrict__ rad_mask,
                                const unsigned char* __restrict__ histo_mask,
                                float* __restrict__ out) {
  const int n_base = blockIdx.x * 16;          // column tile of C
  const int lane   = threadIdx.x;              // wave32
  const int r0     = lane & 15;
  const int khalf  = (lane >> 4) * 2;          // 0 or 2

  const float* aptr0 = feat + (0  + r0) * K_DIM + khalf;   // M-tile 0 (rows 0..15)
  const float* aptr1 = feat + (16 + r0) * K_DIM + khalf;   // M-tile 1 (rows 16..31)
  const float* bptr  = W    + (n_base + r0) * K_DIM + khalf;

  v8f acc0 = {0.f, 0.f, 0.f, 0.f, 0.f, 0.f, 0.f, 0.f};
  v8f acc1 = {0.f, 0.f, 0.f, 0.f, 0.f, 0.f, 0.f, 0.f};

  for (int k = 0; k < K_DIM; k += 16) {
#pragma unroll
    for (int kk = 0; kk < 16; kk += 4) {
      v2f bv = *(const v2f*)(bptr  + k + kk);
      v2f a0 = *(const v2f*)(aptr0 + k + kk);
      v2f a1 = *(const v2f*)(aptr1 + k + kk);
      acc0 = __builtin_amdgcn_wmma_f32_16x16x4_f32(
          false, a0, false, bv, (short)0, acc0, false, false);
      acc1 = __builtin_amdgcn_wmma_f32_16x16x4_f32(
          false, a1, false, bv, (short)0, acc1, false, false);
    }
  }

  // Epilogue: + bias[c] + token[c] * flag[b], flag = !(rad_mask & histo_mask)
  const int   col = n_base + r0;
  const float bs  = bias[col];
  const float tok = token[col];
  const int rsub = (lane >> 4) * 8;            // row offset within tile
#pragma unroll
  for (int i = 0; i < 8; ++i) {
    {
      int row = rsub + i;                      // M-tile 0 -> batch rows 0..15
      float flag = (rad_mask[row] && histo_mask[row]) ? 0.f : 1.f;
      out[row * C_DIM + col] = acc0[i] + bs + tok * flag;
    }
    {
      int row = 16 + rsub + i;                 // M-tile 1 -> batch rows 16..31
      float flag = (rad_mask[row] && histo_mask[row]) ? 0.f : 1.f;
      out[row * C_DIM + col] = acc1[i] + bs + tok * flag;
    }
  }
}

// ---------------------------------------------------------------------------
// Launch
// Inputs: 0 f_rad[32*1024] f32, 1 f_histo[32*2048] f32, 2 rad_mask[32] bool,
//         3 histo_mask[32] bool, 4 W[1024*2048] f32, 5 b[1024] f32,
//         6 token[1024] f32.  Output: emb[32*1024] f32.
// Workspace: stats (96 f32, padded to 1024) + feat (32*2048 f32) = ~266 KB.
// ---------------------------------------------------------------------------
extern "C" void kernel_launch(void* const* d_in, const int* in_sizes, int n_in,
                              void* d_out, int out_size, void* d_ws, size_t ws_size,
                              hipStream_t stream) {
  const float* f_rad   = (const float*)d_in[0];
  const float* f_histo = (const float*)d_in[1];
  const unsigned char* rad_mask   = (const unsigned char*)d_in[2]; // jax bool = 1B
  const unsigned char* histo_mask = (const unsigned char*)d_in[3];
  const float* W     = (const float*)d_in[4];
  const float* bias  = (const float*)d_in[5];
  const float* token = (const float*)d_in[6];
  float* out = (float*)d_out;

  float* stats = (float*)d_ws;          // [96] (T, Btm, inv_nr)
  float* feat  = stats + 1024;          // [32, 2048]

  dpe_stats_kernel<<<BATCH, 256, 0, stream>>>(f_rad, f_histo, stats);
  dpe_feat_kernel<<<(BATCH * I_DIM) / 256, 256, 0, stream>>>(f_rad, stats, feat);
  dpe_gemm_kernel<<<C_DIM / 16, 32, 0, stream>>>(feat, W, bias, token,
                                                 rad_mask, histo_mask, out);
}